// CTCCriterion_19619410608774
// MI455X (gfx1250) — compile-verified
//
#include <hip/hip_runtime.h>
#include <stdint.h>

// ---------------- problem constants (from reference) ----------------
#define BATCH 64
#define NCLS  256
#define TLEN  2048
#define LTGT  128
#define NSTATE 257            // 2L+1
#define ZP   (-1.0e10f)

// ---------------- DP tiling ----------------
#define TCH   128             // t-chunk size
#define NCHUNK (TLEN / TCH)   // 16
#define NROWS 129             // blank row + 128 target rows
#define RSTR  129             // row stride in dwords (128 data + 1 pad) -> 2-way LDS bank conflicts max
#define ROWS_PER_DESC 16
#define NDESC 9               // ceil(129/16)
#define BUFW  (NROWS * RSTR)  // 16641 dwords per buffer

#if __has_builtin(__builtin_amdgcn_tensor_load_to_lds) && __has_builtin(__builtin_amdgcn_s_wait_tensorcnt)
#define CTC_TDM 1
#endif

typedef __attribute__((ext_vector_type(4))) unsigned int v4u;
typedef __attribute__((ext_vector_type(8))) int          v8i;
typedef __attribute__((ext_vector_type(4))) int          v4i;

__device__ __forceinline__ float lse2(float x, float y) {
  float m = fmaxf(x, y);
  return m + __logf(__expf(x - m) + __expf(y - m));
}
__device__ __forceinline__ float lse3(float x, float y, float z) {
  float m = fmaxf(fmaxf(x, y), z);
  return m + __logf(__expf(x - m) + __expf(y - m) + __expf(z - m));
}

// Issue async copy of 129 scattered class-rows (TCH floats each) into LDS at bufbase.
// Uses the Tensor Data Mover in gather mode: 16-bit row indices, 16 rows per descriptor,
// pad_amount=1 dword per 128-dword row -> row stride 129 dwords (max 2-way bank conflict).
__device__ __forceinline__ void issue_chunk(const float* xb, int t0,
                                            float* bufbase, const uint32_t* idx32) {
#ifdef CTC_TDM
  uint64_t g = (uint64_t)(uintptr_t)(xb + t0);   // tile start: &x[b][0][t0]
  uint32_t lbase = (uint32_t)(uintptr_t)bufbase; // LDS byte offset (low 32 bits of flat addr)
  #pragma unroll
  for (int d = 0; d < NDESC; ++d) {
    uint32_t nidx = (d == NDESC - 1) ? 1u : 16u;       // gather-row count (tile_dim1)
    v4u g0;
    g0[0] = 1u | (1u << 31);                           // count=1, gather_mode=1, 16-bit indices
    g0[1] = lbase + (uint32_t)(d * ROWS_PER_DESC * RSTR * 4);
    g0[2] = (uint32_t)g;                               // global_addr[31:0]
    g0[3] = ((uint32_t)(g >> 32) & 0x01FFFFFFu) | (2u << 30); // addr[56:32], type=2
    v8i g1;
    g1[0] = (int)((2u << 16) | (1u << 20) | (6u << 22)); // data_size=4B, pad_en, interval=128dw, amount=1dw
    g1[1] = (int)(((uint32_t)TLEN & 0xFFFFu) << 16);     // tensor_dim0 lo16
    g1[2] = (int)(((uint32_t)NCLS & 0xFFFFu) << 16);     // tensor_dim0 hi=0 | tensor_dim1 lo16
    g1[3] = (int)(((uint32_t)TCH  & 0xFFFFu) << 16);     // tensor_dim1 hi=0 | tile_dim0=128
    g1[4] = (int)nidx;                                   // tile_dim1 = #valid gather indices
    g1[5] = TLEN;                                        // tensor_dim0_stride lo32
    g1[6] = 0; g1[7] = 0;
    v4i g2, g3;                                          // packed 16-bit row (class) indices
    g2[0] = (int)idx32[d * 8 + 0]; g2[1] = (int)idx32[d * 8 + 1];
    g2[2] = (int)idx32[d * 8 + 2]; g2[3] = (int)idx32[d * 8 + 3];
    g3[0] = (int)idx32[d * 8 + 4]; g3[1] = (int)idx32[d * 8 + 5];
    g3[2] = (int)idx32[d * 8 + 6]; g3[3] = (int)idx32[d * 8 + 7];
    v8i g4 = {0, 0, 0, 0, 0, 0, 0, 0};                   // extra descriptor payload (unused)
    __builtin_amdgcn_tensor_load_to_lds(g0, g1, g2, g3, g4, 0);
  }
#else
  // synchronous fallback: wave0 copies rows itself
  const int lane = (int)(threadIdx.x & 31u);
  for (int r = 0; r < NROWS; ++r) {
    uint32_t cls = (idx32[r >> 1] >> ((r & 1) * 16)) & 0xFFFFu;
    const float* src = xb + (size_t)cls * TLEN + t0;
    float* dst = bufbase + r * RSTR;
    #pragma unroll
    for (int j = 0; j < 4; ++j) dst[lane + 32 * j] = src[lane + 32 * j];
  }
#endif
}

__device__ __forceinline__ void wait_emit() {
#ifdef CTC_TDM
  __builtin_amdgcn_s_wait_tensorcnt(0);
#endif
}

__global__ __launch_bounds__(256) void ctc_fwd_kernel(const float* __restrict__ x,
                                                      const int* __restrict__ targets,
                                                      float* __restrict__ out) {
  __shared__ float    lse_sh[TLEN];       // 8 KB: log-softmax normalizer per t
  __shared__ uint32_t idx32_sh[72];       // packed 16-bit gather indices (129 valid, zero-padded)
  __shared__ float    emit_sh[2 * BUFW];  // ~130 KB: double-buffered gathered class rows

  const int b   = (int)blockIdx.x;
  const int tid = (int)threadIdx.x;
  const float* xb = x + (size_t)b * NCLS * TLEN;

  // ---------- phase 1: streaming online-softmax over classes (all 256 threads) ----------
  {
    float m[8], s[8];
    #pragma unroll
    for (int j = 0; j < 8; ++j) { m[j] = -1.0e30f; s[j] = 0.0f; }
    const int tbase = tid * 8;
    for (int c = 0; c < NCLS; ++c) {
      const float4* pp = (const float4*)(xb + (size_t)c * TLEN + tbase);
      float4 v0 = pp[0], v1 = pp[1];
      float vv[8] = {v0.x, v0.y, v0.z, v0.w, v1.x, v1.y, v1.z, v1.w};
      #pragma unroll
      for (int j = 0; j < 8; ++j) {
        float mn = fmaxf(m[j], vv[j]);
        s[j] = s[j] * __expf(m[j] - mn) + __expf(vv[j] - mn);
        m[j] = mn;
      }
    }
    #pragma unroll
    for (int j = 0; j < 8; ++j) lse_sh[tbase + j] = m[j] + __logf(s[j]);
  }

  // gather-index table: row 0 = blank class 0; row 1+k = targets[b][k]
  if (tid < 72) {
    int i0 = 2 * tid, i1 = 2 * tid + 1;
    uint32_t lo = 0, hi = 0;
    if (i0 > 0 && i0 < NROWS) lo = (uint32_t)targets[b * LTGT + (i0 - 1)] & 0xFFFFu;
    if (i1 < NROWS)           hi = (uint32_t)targets[b * LTGT + (i1 - 1)] & 0xFFFFu;
    idx32_sh[tid] = lo | (hi << 16);
  }
  __syncthreads();
  if (tid >= 32) return;   // phase 2 is a single wave (no barriers in the serial chain)

  // ---------- phase 2: forward CTC DP, one wave32, 8-9 states per lane ----------
  const int lane = tid;
  const int* tgt = targets + b * LTGT;
  bool z[4];                       // s2-mask for odd states: k==0 or repeated label
  #pragma unroll
  for (int j = 0; j < 4; ++j) {
    int k = 4 * lane + j;
    z[j] = (k == 0) ? true : (tgt[k] == tgt[k - 1]);
  }
  float a[9];
  #pragma unroll
  for (int i = 0; i < 9; ++i) a[i] = ZP;
  if (lane == 0) a[0] = 0.0f;      // init: alpha[0] = 0, rest ZP

  issue_chunk(xb, 0, emit_sh, idx32_sh);
  wait_emit();

  const int r1 = 4 * lane + 1;     // first odd-state row for this lane
  for (int ch = 0; ch < NCHUNK; ++ch) {
    if (ch + 1 < NCHUNK)
      issue_chunk(xb, (ch + 1) * TCH, emit_sh + ((ch + 1) & 1) * BUFW, idx32_sh);
    const float* buf = emit_sh + (ch & 1) * BUFW;

    for (int tc = 0; tc < TCH; ++tc) {
      int t = ch * TCH + tc;
      float lt = lse_sh[t];
      float eB = buf[tc] - lt;                        // blank emission (row 0, broadcast)
      float e1 = buf[(r1 + 0) * RSTR + tc] - lt;
      float e3 = buf[(r1 + 1) * RSTR + tc] - lt;
      float e5 = buf[(r1 + 2) * RSTR + tc] - lt;
      float e7 = buf[(r1 + 3) * RSTR + tc] - lt;
      float p1 = __shfl_up(a[7], 1u);                 // alpha[p-1] across lane boundary
      float p2 = __shfl_up(a[6], 1u);                 // alpha[p-2] across lane boundary
      if (lane == 0) { p1 = ZP; p2 = ZP; }
      float n0 = lse2(a[0], p1) + eB;                 // even states: blank, s2 always masked
      float n1 = lse3(a[1], a[0], z[0] ? ZP : p2) + e1;
      float n2 = lse2(a[2], a[1]) + eB;
      float n3 = lse3(a[3], a[2], z[1] ? ZP : a[1]) + e3;
      float n4 = lse2(a[4], a[3]) + eB;
      float n5 = lse3(a[5], a[4], z[2] ? ZP : a[3]) + e5;
      float n6 = lse2(a[6], a[5]) + eB;
      float n7 = lse3(a[7], a[6], z[3] ? ZP : a[5]) + e7;
      float n8 = lse2(a[8], a[7]) + eB;               // state 256 (only lane 31's is real)
      a[0] = n0; a[1] = n1; a[2] = n2; a[3] = n3; a[4] = n4;
      a[5] = n5; a[6] = n6; a[7] = n7; a[8] = n8;
    }
    wait_emit();   // prefetch for next chunk has landed
  }

  // loss = -logaddexp(alpha[T-1][P-1], alpha[T-1][P-2]) (== -logsumexp(combined[0]))
  if (lane == 31) out[b] = -lse2(a[7], a[8]);
}

extern "C" void kernel_launch(void* const* d_in, const int* in_sizes, int n_in,
                              void* d_out, int out_size, void* d_ws, size_t ws_size,
                              hipStream_t stream) {
  (void)in_sizes; (void)n_in; (void)out_size; (void)d_ws; (void)ws_size;
  const float* x       = (const float*)d_in[0];
  const int*   targets = (const int*)d_in[1];
  float*       out     = (float*)d_out;
  ctc_fwd_kernel<<<dim3(BATCH), dim3(256), 0, stream>>>(x, targets, out);
}